// Triplane_69879117906796
// MI455X (gfx1250) — compile-verified
//
#include <hip/hip_runtime.h>
#include <hip/hip_bf16.h>
#include <cmath>

#define CH 16
#define BLOCK 256

typedef float v4f __attribute__((ext_vector_type(4)));

// ---- LDS byte-offset of a __shared__ object (addrspace(3) ptrtoint) ----
typedef __attribute__((address_space(3))) const void lds_cv;
static __device__ __forceinline__ unsigned lds_offset(const void* p) {
  return (unsigned)(size_t)(lds_cv*)p;
}

// ---- transpose [3,16,res,res] -> [3,res,res,16] (channel-last) ----
// grid: (ceil(res/BLOCK), res, 3) -> no integer division anywhere
__global__ __launch_bounds__(BLOCK) void transpose_chlast(
    const float* __restrict__ fm, float* __restrict__ dst, int res) {
  const int c = blockIdx.x * BLOCK + threadIdx.x;
  if (c >= res) return;
  const int r = blockIdx.y;
  const int p = blockIdx.z;
  const size_t plane = (size_t)res * res;
  const float* src = fm + ((size_t)p * CH) * plane + (size_t)r * res + c;
  float v[CH];
#pragma unroll
  for (int ch = 0; ch < CH; ++ch) v[ch] = src[(size_t)ch * plane];
  v4f* o = (v4f*)(dst + (((size_t)p * res + r) * res + c) * CH);
#pragma unroll
  for (int j = 0; j < 4; ++j) {
    v4f w = {v[4 * j + 0], v[4 * j + 1], v[4 * j + 2], v[4 * j + 3]};
    o[j] = w;
  }
}

// ---- bilinear tap for one plane, exactly matching the reference math ----
template <bool CHLAST>
static __device__ __forceinline__ void eval_plane(
    const float* __restrict__ fm, const float* __restrict__ fmT, int p,
    float u, float su, float v, float sv, int D, float* __restrict__ so) {
  float x = ((u / su) + 1.0f) * (float)(D - 1) * 0.5f;
  float y = ((v / sv) + 1.0f) * (float)(D - 1) * 0.5f;
  float fx = floorf(x), fy = floorf(y);
  float tx = x - fx, ty = y - fy;
  float txc = (fx + 1.0f) - x, tyc = (fy + 1.0f) - y;
  float wa = txc * tyc, wb = tx * tyc, wc = txc * ty, wd = tx * ty;
  int r0 = min(max((int)fx, 0), D - 2);
  int r1 = min(max((int)fx + 1, 1), D - 1);
  int c0 = min(max((int)fy, 0), D - 2);
  int c1 = min(max((int)fy + 1, 1), D - 1);
  if constexpr (CHLAST) {
    const v4f* A  = (const v4f*)(fmT + (((size_t)p * D + r0) * D + c0) * CH);
    const v4f* B  = (const v4f*)(fmT + (((size_t)p * D + r1) * D + c0) * CH);
    const v4f* C  = (const v4f*)(fmT + (((size_t)p * D + r0) * D + c1) * CH);
    const v4f* Dd = (const v4f*)(fmT + (((size_t)p * D + r1) * D + c1) * CH);
#pragma unroll
    for (int j = 0; j < 4; ++j) {
      v4f a = A[j], b = B[j], c = C[j], d = Dd[j];
      v4f o = wa * a + wb * b + wc * c + wd * d;
      *(v4f*)(so + j * 4) = o;  // 16B-aligned LDS store
    }
  } else {
    const size_t plane = (size_t)D * D;
#pragma unroll
    for (int ch = 0; ch < CH; ++ch) {
      const float* base = fm + ((size_t)(p * CH + ch)) * plane;
      const float* rp0 = base + (size_t)r0 * D;
      const float* rp1 = base + (size_t)r1 * D;
      so[ch] = wa * rp0[c0] + wb * rp1[c0] + wc * rp0[c1] + wd * rp1[c1];
    }
  }
}

template <bool CHLAST>
__global__ __launch_bounds__(BLOCK) void triplane_kernel(
    const float* __restrict__ xyz, const float* __restrict__ fm,
    const float* __restrict__ fmT, float* __restrict__ out,
    long long N, int res) {
  __shared__ float sxyz[BLOCK * 3];
  __shared__ float sout[BLOCK * 48];
  const int tid = threadIdx.x;
  const long long blockStart = (long long)blockIdx.x * BLOCK;

  // --- Stage this block's xyz into LDS via CDNA5 async copy (ASYNCcnt) ---
  {
    const long long totalDw = N * 3;
    const long long baseDw = blockStart * 3;
    if (tid < (BLOCK * 3) / 4) {  // 192 lanes x 16B = 3 KB
      if (baseDw + (long long)tid * 4 + 4 <= totalDw) {
        unsigned loff = lds_offset(sxyz) + (unsigned)tid * 16u;
        unsigned goff = (unsigned)tid * 16u;
        const float* sbase = xyz + baseDw;
        asm volatile("global_load_async_to_lds_b128 %0, %1, %2"
                     :: "v"(loff), "v"(goff), "s"(sbase) : "memory");
      } else {  // ragged tail (N % BLOCK != 0): plain path
#pragma unroll
        for (int k = 0; k < 4; ++k) {
          long long g = baseDw + (long long)tid * 4 + k;
          if (g < totalDw) sxyz[tid * 4 + k] = xyz[g];
        }
      }
    }
#if __has_builtin(__builtin_amdgcn_s_wait_asynccnt)
    __builtin_amdgcn_s_wait_asynccnt(0);
#else
    asm volatile("s_wait_asynccnt 0x0" ::: "memory");
#endif
    __syncthreads();
  }

  // --- per-point bilinear gather on 3 planes ---
  const long long pidx = blockStart + tid;
  if (pidx < N) {
    const float px = sxyz[tid * 3 + 0];
    const float py = sxyz[tid * 3 + 1];
    const float pz = sxyz[tid * 3 + 2];
    float* so = sout + tid * 48;
    eval_plane<CHLAST>(fm, fmT, 0, px, 1.0f, py, 1.0f, res, so + 0);   // f_xy
    eval_plane<CHLAST>(fm, fmT, 2, py, 1.0f, pz, 0.05f, res, so + 16); // f_yz
    eval_plane<CHLAST>(fm, fmT, 1, pz, 0.05f, px, 1.0f, res, so + 32); // f_zx
  }
  __syncthreads();

  // --- coalesced non-temporal stores: each wave streams its 6 KB region ---
  {
    const int lane = tid & 31;
    const int wv = tid >> 5;
    const float* src = sout + wv * (32 * 48);
    const size_t gbase = (size_t)blockStart * 48 + (size_t)wv * (32 * 48);
    const size_t limit = (size_t)N * 48;
#pragma unroll
    for (int i = 0; i < 12; ++i) {
      size_t off = gbase + (size_t)i * 128 + (size_t)lane * 4;
      if (off + 4 <= limit) {
        v4f vv = *(const v4f*)(src + i * 128 + lane * 4);
        __builtin_nontemporal_store(vv, (v4f*)(out + off));
      }
    }
  }
}

extern "C" void kernel_launch(void* const* d_in, const int* in_sizes, int n_in,
                              void* d_out, int out_size, void* d_ws, size_t ws_size,
                              hipStream_t stream) {
  (void)n_in; (void)out_size;
  const float* xyz = (const float*)d_in[0];
  const float* fm  = (const float*)d_in[1];
  float* out = (float*)d_out;

  long long N = (long long)in_sizes[0] / 3;
  long long planeElems = (long long)in_sizes[1];
  int res = (int)llround(sqrt((double)(planeElems / (3 * CH))));
  int blocks = (int)((N + BLOCK - 1) / BLOCK);
  size_t needT = (size_t)planeElems * sizeof(float);

  if (d_ws && ws_size >= needT) {
    dim3 tg((unsigned)((res + BLOCK - 1) / BLOCK), (unsigned)res, 3u);
    transpose_chlast<<<tg, BLOCK, 0, stream>>>(fm, (float*)d_ws, res);
    triplane_kernel<true><<<blocks, BLOCK, 0, stream>>>(
        xyz, fm, (const float*)d_ws, out, N, res);
  } else {
    triplane_kernel<false><<<blocks, BLOCK, 0, stream>>>(
        xyz, fm, nullptr, out, N, res);
  }
}